// GNNDecoder_73023033967407
// MI455X (gfx1250) — compile-verified
//
#include <hip/hip_runtime.h>

typedef __attribute__((ext_vector_type(2))) float v2f;
typedef __attribute__((ext_vector_type(8))) float v8f;

#define EMB   128
#define EMB2  256
#define KEA   12     // 9 edge attrs + 1 count + 2 zero pad (K multiple of 4)
#define BN_EPS 1e-5f

// ---------------------------------------------------------------------------
// GEMM 1: h = PReLU(x) @ W_enc        [N,128] x [128,128]
// block = 128 threads (4 waves). Block covers 16 rows; wave w covers cols
// [32w, 32w+32) as two 16x16 WMMA tiles. K-loop in steps of 4 (f32 WMMA).
// ---------------------------------------------------------------------------
__global__ __launch_bounds__(128) void k_h(const float* __restrict__ x,
                                           const float* __restrict__ Wenc,
                                           const float* __restrict__ prelu_a,
                                           float* __restrict__ h) {
  const int lane = threadIdx.x & 31;
  const int wave = threadIdx.x >> 5;
  const int rowbase = blockIdx.x * 16;
  const int m    = lane & 15;          // A row / B col within tile
  const int koff = (lane >> 4) * 2;    // K sub-offset per half-wave
  const int colw = wave * 32;
  const float aw = prelu_a[0];

  const float* xr = x + (size_t)(rowbase + m) * EMB;
  v8f acc0 = (v8f)0.0f, acc1 = (v8f)0.0f;

  for (int k = 0; k < EMB; k += 4) {
    float a0 = xr[k + koff];
    float a1 = xr[k + koff + 1];
    a0 = (a0 >= 0.0f) ? a0 : a0 * aw;
    a1 = (a1 >= 0.0f) ? a1 : a1 * aw;
    v2f a; a.x = a0; a.y = a1;
    const float* w0 = Wenc + (size_t)(k + koff) * EMB + colw + m;
    v2f b0; b0.x = w0[0];  b0.y = w0[EMB];
    v2f b1; b1.x = w0[16]; b1.y = w0[EMB + 16];
    acc0 = __builtin_amdgcn_wmma_f32_16x16x4_f32(false, a, false, b0, (short)0, acc0, false, false);
    acc1 = __builtin_amdgcn_wmma_f32_16x16x4_f32(false, a, false, b1, (short)0, acc1, false, false);
  }

  const int rofs = (lane >> 4) * 8;    // C/D layout: VGPR i -> row i + 8*(lane/16)
#pragma unroll
  for (int i = 0; i < 8; ++i) {
    const size_t r = (size_t)(rowbase + rofs + i) * EMB;
    h[r + colw + m]      = acc0[i];
    h[r + colw + 16 + m] = acc1[i];
  }
}

// ---------------------------------------------------------------------------
// Zero masked node rows of h
// ---------------------------------------------------------------------------
__global__ void k_mask(const long long* __restrict__ idx, float* __restrict__ h,
                       int nmask) {
  int b = blockIdx.x;
  if (b < nmask) {
    long long n = idx[b];
    h[(size_t)n * EMB + threadIdx.x] = 0.0f;
  }
}

// aggr_h = h  (self-loop contribution: src == dst adds h[n] exactly once)
__global__ void k_copy(const float* __restrict__ a, float* __restrict__ b, size_t n) {
  size_t i = (size_t)blockIdx.x * 256 + threadIdx.x;
  if (i < n) b[i] = a[i];
}

// aggr_ea init: self loop attr is one-hot at 7, count (comp 9) = 1, pad = 0
__global__ void k_eainit(float* __restrict__ aggr_ea, int total) {
  int i = blockIdx.x * 256 + threadIdx.x;
  if (i < total) {
    int c = i % KEA;
    aggr_ea[i] = (c == 7 || c == 9) ? 1.0f : 0.0f;
  }
}

// ---------------------------------------------------------------------------
// Edge scatter: one wave per edge. 32 lanes x float4 gather of h[src],
// f32 atomics into aggr_h[dst] (stays in 192MB L2). Lanes 0..9 scatter the
// 9 edge attrs + count into aggr_ea[dst].
// ---------------------------------------------------------------------------
__global__ __launch_bounds__(256) void k_edge(const long long* __restrict__ ei,
                                              const float* __restrict__ ea,
                                              const float* __restrict__ h,
                                              float* __restrict__ aggr_h,
                                              float* __restrict__ aggr_ea,
                                              int E) {
  int e = blockIdx.x * 8 + (threadIdx.x >> 5);
  if (e >= E) return;
  int lane = threadIdx.x & 31;
  long long src = ei[e];
  long long dst = ei[(size_t)E + e];
  const float4* hp = reinterpret_cast<const float4*>(h + (size_t)src * EMB);
  float4 v = hp[lane];
  float* dp = aggr_h + (size_t)dst * EMB + lane * 4;
  atomicAdd(dp + 0, v.x);
  atomicAdd(dp + 1, v.y);
  atomicAdd(dp + 2, v.z);
  atomicAdd(dp + 3, v.w);
  if (lane < 9)
    atomicAdd(aggr_ea + (size_t)dst * KEA + lane, ea[(size_t)e * 9 + lane]);
  else if (lane == 9)
    atomicAdd(aggr_ea + (size_t)dst * KEA + 9, 1.0f);
}

// ---------------------------------------------------------------------------
// Wcomb[12,256] = [W_e ; b_e ; 0 ; 0] (12x128) @ W1[128:256, :]
// (folds the linear edge encoder through the bottom half of W1)
// ---------------------------------------------------------------------------
__global__ __launch_bounds__(256) void k_wcomb(const float* __restrict__ We,
                                               const float* __restrict__ be,
                                               const float* __restrict__ W1,
                                               float* __restrict__ Wc) {
  int j = threadIdx.x;  // output column 0..255
  for (int i = 0; i < KEA; ++i) {
    float s = 0.0f;
    if (i < 9) {
      for (int c = 0; c < EMB; ++c)
        s += We[i * EMB + c] * W1[(size_t)(EMB + c) * EMB2 + j];
    } else if (i == 9) {
      for (int c = 0; c < EMB; ++c)
        s += be[c] * W1[(size_t)(EMB + c) * EMB2 + j];
    }
    Wc[i * EMB2 + j] = s;
  }
}

// ---------------------------------------------------------------------------
// GEMM 2: z = aggr_h @ W1[0:128,:] + aggr_ea12 @ Wcomb + b1   [N,256]
// block covers 16 rows; wave w covers cols [64w, 64w+64) -> 4 WMMA tiles.
// ---------------------------------------------------------------------------
__global__ __launch_bounds__(128) void k_z(const float* __restrict__ aggr_h,
                                           const float* __restrict__ aggr_ea,
                                           const float* __restrict__ W1,
                                           const float* __restrict__ Wc,
                                           const float* __restrict__ b1,
                                           float* __restrict__ z) {
  const int lane = threadIdx.x & 31;
  const int wave = threadIdx.x >> 5;
  const int rowbase = blockIdx.x * 16;
  const int m    = lane & 15;
  const int koff = (lane >> 4) * 2;
  const int colw = wave * 64;

  v8f acc[4] = {};
  const float* ar = aggr_h + (size_t)(rowbase + m) * EMB;
  for (int k = 0; k < EMB; k += 4) {
    v2f a; a.x = ar[k + koff]; a.y = ar[k + koff + 1];
    const float* w0 = W1 + (size_t)(k + koff) * EMB2 + colw + m;
#pragma unroll
    for (int ct = 0; ct < 4; ++ct) {
      v2f b; b.x = w0[ct * 16]; b.y = w0[EMB2 + ct * 16];
      acc[ct] = __builtin_amdgcn_wmma_f32_16x16x4_f32(false, a, false, b, (short)0, acc[ct], false, false);
    }
  }
  const float* er = aggr_ea + (size_t)(rowbase + m) * KEA;
  for (int k = 0; k < KEA; k += 4) {
    v2f a; a.x = er[k + koff]; a.y = er[k + koff + 1];
    const float* w0 = Wc + (size_t)(k + koff) * EMB2 + colw + m;
#pragma unroll
    for (int ct = 0; ct < 4; ++ct) {
      v2f b; b.x = w0[ct * 16]; b.y = w0[EMB2 + ct * 16];
      acc[ct] = __builtin_amdgcn_wmma_f32_16x16x4_f32(false, a, false, b, (short)0, acc[ct], false, false);
    }
  }

  const int rofs = (lane >> 4) * 8;
#pragma unroll
  for (int ct = 0; ct < 4; ++ct) {
    const int col = colw + ct * 16 + m;
    const float bias = b1[col];
#pragma unroll
    for (int i = 0; i < 8; ++i)
      z[(size_t)(rowbase + rofs + i) * EMB2 + col] = acc[ct][i] + bias;
  }
}

// ---------------------------------------------------------------------------
// BatchNorm training stats: per-column sum / sumsq over all N rows
// ---------------------------------------------------------------------------
__global__ void k_zero2(float* __restrict__ a, float* __restrict__ b) {
  a[threadIdx.x] = 0.0f;
  b[threadIdx.x] = 0.0f;
}

__global__ __launch_bounds__(256) void k_stats(const float* __restrict__ z,
                                               float* __restrict__ colsum,
                                               float* __restrict__ colsq, int N) {
  int t  = threadIdx.x;                 // column
  int r0 = blockIdx.x * 256;
  int r1 = min(r0 + 256, N);
  float s = 0.0f, q = 0.0f;
  for (int r = r0; r < r1; ++r) {
    float v = z[(size_t)r * EMB2 + t];
    s += v;
    q += v * v;
  }
  atomicAdd(colsum + t, s);
  atomicAdd(colsq + t, q);
}

__global__ void k_bn(const float* __restrict__ colsum, const float* __restrict__ colsq,
                     const float* __restrict__ gamma, const float* __restrict__ beta,
                     float* __restrict__ scale, float* __restrict__ shift, int N) {
  int t = threadIdx.x;
  float invN = 1.0f / (float)N;
  float mean = colsum[t] * invN;
  float var  = colsq[t] * invN - mean * mean;       // biased var (jnp.var)
  float sc   = gamma[t] * rsqrtf(var + BN_EPS);
  scale[t] = sc;
  shift[t] = beta[t] - mean * sc;
}

// ---------------------------------------------------------------------------
// GEMM 3: out = relu(z*scale + shift) @ W2 + b2   [N,256] x [256,128]
// BN affine + ReLU fused into the A-fragment load.
// ---------------------------------------------------------------------------
__global__ __launch_bounds__(128) void k_out(const float* __restrict__ z,
                                             const float* __restrict__ W2,
                                             const float* __restrict__ scale,
                                             const float* __restrict__ shift,
                                             const float* __restrict__ b2,
                                             float* __restrict__ out) {
  const int lane = threadIdx.x & 31;
  const int wave = threadIdx.x >> 5;
  const int rowbase = blockIdx.x * 16;
  const int m    = lane & 15;
  const int koff = (lane >> 4) * 2;
  const int colw = wave * 32;

  const float* zr = z + (size_t)(rowbase + m) * EMB2;
  v8f acc0 = (v8f)0.0f, acc1 = (v8f)0.0f;

  for (int k = 0; k < EMB2; k += 4) {
    const int kk = k + koff;
    float a0 = fmaxf(zr[kk]     * scale[kk]     + shift[kk],     0.0f);
    float a1 = fmaxf(zr[kk + 1] * scale[kk + 1] + shift[kk + 1], 0.0f);
    v2f a; a.x = a0; a.y = a1;
    const float* w0 = W2 + (size_t)kk * EMB + colw + m;
    v2f b0; b0.x = w0[0];  b0.y = w0[EMB];
    v2f b1; b1.x = w0[16]; b1.y = w0[EMB + 16];
    acc0 = __builtin_amdgcn_wmma_f32_16x16x4_f32(false, a, false, b0, (short)0, acc0, false, false);
    acc1 = __builtin_amdgcn_wmma_f32_16x16x4_f32(false, a, false, b1, (short)0, acc1, false, false);
  }

  const int rofs = (lane >> 4) * 8;
  const float bb0 = b2[colw + m];
  const float bb1 = b2[colw + 16 + m];
#pragma unroll
  for (int i = 0; i < 8; ++i) {
    const size_t r = (size_t)(rowbase + rofs + i) * EMB;
    out[r + colw + m]      = acc0[i] + bb0;
    out[r + colw + 16 + m] = acc1[i] + bb1;
  }
}

// ---------------------------------------------------------------------------
extern "C" void kernel_launch(void* const* d_in, const int* in_sizes, int n_in,
                              void* d_out, int out_size, void* d_ws, size_t ws_size,
                              hipStream_t stream) {
  const float*     x       = (const float*)d_in[0];
  const long long* ei      = (const long long*)d_in[1];
  const float*     ea      = (const float*)d_in[2];
  const long long* mask    = (const long long*)d_in[3];
  const float*     prelu_a = (const float*)d_in[4];
  const float*     Wenc    = (const float*)d_in[5];
  const float*     We      = (const float*)d_in[6];
  const float*     be      = (const float*)d_in[7];
  const float*     W1      = (const float*)d_in[8];
  const float*     b1      = (const float*)d_in[9];
  const float*     gamma   = (const float*)d_in[10];
  const float*     beta    = (const float*)d_in[11];
  const float*     W2      = (const float*)d_in[12];
  const float*     b2      = (const float*)d_in[13];

  const int N     = in_sizes[0] / EMB;   // 50000 (multiple of 16)
  const int E     = in_sizes[1] / 2;     // 800000
  const int nmask = in_sizes[3];         // 5000

  // workspace layout
  char* ws = (char*)d_ws;
  float* h       = (float*)ws; ws += (size_t)N * EMB  * sizeof(float);
  float* aggr_h  = (float*)ws; ws += (size_t)N * EMB  * sizeof(float);
  float* aggr_ea = (float*)ws; ws += (size_t)N * KEA  * sizeof(float);
  float* z       = (float*)ws; ws += (size_t)N * EMB2 * sizeof(float);
  float* Wc      = (float*)ws; ws += (size_t)KEA * EMB2 * sizeof(float);
  float* colsum  = (float*)ws; ws += 256 * sizeof(float);
  float* colsq   = (float*)ws; ws += 256 * sizeof(float);
  float* bnscale = (float*)ws; ws += 256 * sizeof(float);
  float* bnshift = (float*)ws; ws += 256 * sizeof(float);

  const int nrb = N / 16;  // 3125 row blocks, exact

  // 1) h = PReLU(x) @ W_enc ; zero masked rows
  k_h<<<nrb, 128, 0, stream>>>(x, Wenc, prelu_a, h);
  k_mask<<<nmask, EMB, 0, stream>>>(mask, h, nmask);

  // 2) init aggregation with self-loop contributions
  k_copy<<<(int)(((size_t)N * EMB + 255) / 256), 256, 0, stream>>>(h, aggr_h, (size_t)N * EMB);
  k_eainit<<<(N * KEA + 255) / 256, 256, 0, stream>>>(aggr_ea, N * KEA);

  // 3) edge scatter (gather h[src], atomic-add into dst; 10-dim edge attrs)
  k_edge<<<(E + 7) / 8, 256, 0, stream>>>(ei, ea, h, aggr_h, aggr_ea, E);

  // 4) fold edge encoder through bottom half of W1
  k_wcomb<<<1, 256, 0, stream>>>(We, be, W1, Wc);

  // 5) z = aggr @ W1 + b1 (via split K: 128 WMMA + 12 WMMA)
  k_z<<<nrb, 128, 0, stream>>>(aggr_h, aggr_ea, W1, Wc, b1, z);

  // 6) BN training statistics
  k_zero2<<<1, 256, 0, stream>>>(colsum, colsq);
  k_stats<<<(N + 255) / 256, 256, 0, stream>>>(z, colsum, colsq, N);
  k_bn<<<1, 256, 0, stream>>>(colsum, colsq, gamma, beta, bnscale, bnshift, N);

  // 7) out = relu(bn(z)) @ W2 + b2 (BN+ReLU fused into A-operand load)
  k_out<<<nrb, 128, 0, stream>>>(z, W2, bnscale, bnshift, b2, (float*)d_out);
}